// DifferentiableRasterizer_32031866093904
// MI455X (gfx1250) — compile-verified
//
#include <hip/hip_runtime.h>
#include <hip/hip_bf16.h>

typedef __attribute__((ext_vector_type(2))) float v2f;
typedef __attribute__((ext_vector_type(8))) float v8f;

#define B_ 2
#define V_ 1024
#define F_ 2048
#define C_ 3
#define H_ 128
#define W_ 128
#define NFG_ (F_ / 4)   // face groups of 4 faces (16 planes) each

__device__ __constant__ const float kFTINY = 1.1754944e-35f;   // tiny*1000
__device__ __constant__ const float kINF   = 3.4028235e35f;    // max*0.001
__device__ __constant__ const float kLINF  = 3.4028235e34f;    // max*0.0001

// Barycentric plane coefficients, matching _bary_coeffs in the reference.
// l[0..2]=l0{x,y,c}, l[3..5]=l1{x,y,c}, l[6..8]=l2{x,y,c}
__device__ __forceinline__ void bary_coeffs(float x0, float x1, float x2,
                                            float y0, float y1, float y2,
                                            float* __restrict__ l) {
  float det = (y1 - y2) * (x0 - x2) + (x2 - x1) * (y0 - y2);
  float dets = copysignf(fmaxf(fabsf(det), kFTINY), det);
  float inv = 1.0f / dets;
  float l0x = (y1 - y2) * inv, l0y = (x2 - x1) * inv;
  float l0c = -l0x * x2 - l0y * y2;
  float l1x = (y2 - y0) * inv, l1y = (x0 - x2) * inv;
  float l1c = -l1x * x2 - l1y * y2;
  l[0] = l0x; l[1] = l0y; l[2] = l0c;
  l[3] = l1x; l[4] = l1y; l[5] = l1c;
  l[6] = -l0x - l1x; l[7] = -l0y - l1y; l[8] = 1.0f - l0c - l1c;
}

// Phase 1: per-face plane setup + culling. Output: 4 planes * (cx,cy,cc,0) per face.
__global__ void plane_setup(const float* __restrict__ pt2d, const float* __restrict__ pt3d,
                            const float* __restrict__ normal, const float* __restrict__ Rm,
                            const float* __restrict__ Tm, const int* __restrict__ face,
                            float* __restrict__ planes) {
  int t = blockIdx.x * blockDim.x + threadIdx.x;
  if (t >= B_ * F_) return;
  int b = t / F_, f = t - b * F_;
  int f0 = face[f], f1 = face[F_ + f], f2 = face[2 * F_ + f];

  const float* p2 = pt2d + b * 3 * V_;
  float x0 = p2[f0],          x1 = p2[f1],          x2 = p2[f2];
  float y0 = p2[V_ + f0],     y1 = p2[V_ + f1],     y2 = p2[V_ + f2];
  float z0 = p2[2 * V_ + f0], z1 = p2[2 * V_ + f1], z2 = p2[2 * V_ + f2];

  // offs = R^T @ T
  const float* R = Rm + b * 9;
  const float* T = Tm + b * 3;
  float o0 = R[0] * T[0] + R[3] * T[1] + R[6] * T[2];
  float o1 = R[1] * T[0] + R[4] * T[1] + R[7] * T[2];
  float o2 = R[2] * T[0] + R[5] * T[1] + R[8] * T[2];

  const float* p3 = pt3d + b * 3 * V_;
  const float* nr = normal + b * 3 * F_;
  float dotv = (p3[f0] + o0) * nr[f] + (p3[V_ + f0] + o1) * nr[F_ + f] +
               (p3[2 * V_ + f0] + o2) * nr[2 * F_ + f];
  bool valid = (dotv < 0.0f) && (fminf(z0, fminf(z1, z2)) > 0.0f);

  float l[9];
  bary_coeffs(x0, x1, x2, y0, y1, y2, l);
  float Dx = z0 * l[0] + z1 * l[3] + z2 * l[6];
  float Dy = z0 * l[1] + z1 * l[4] + z2 * l[7];
  float Dc = z0 * l[2] + z1 * l[5] + z2 * l[8];

  float4* out = (float4*)(planes + (size_t)t * 16);
  if (valid) {
    out[0] = make_float4(l[0], l[1], l[2], 0.0f);
    out[1] = make_float4(l[3], l[4], l[5], 0.0f);
    out[2] = make_float4(l[6], l[7], l[8], 0.0f);
    out[3] = make_float4(Dx, Dy, Dc, 0.0f);
  } else {
    out[0] = make_float4(0.0f, 0.0f, -1.0f, 0.0f);  // l0 == -1 -> never covered
    out[1] = make_float4(0.0f, 0.0f, 0.0f, 0.0f);
    out[2] = make_float4(0.0f, 0.0f, 0.0f, 0.0f);
    out[3] = make_float4(0.0f, 0.0f, kINF, 0.0f);
  }
}

__device__ __forceinline__ void shade_pixel(int b, int pix, float bestd, int besti,
                                            const float* __restrict__ pt2d,
                                            const float* __restrict__ color,
                                            const int* __restrict__ face,
                                            float* __restrict__ out) {
  float m = (bestd < kLINF) ? 1.0f : 0.0f;
  int f0 = face[besti], f1 = face[F_ + besti], f2 = face[2 * F_ + besti];
  const float* p2 = pt2d + b * 3 * V_;
  float x0 = p2[f0],      x1 = p2[f1],      x2 = p2[f2];
  float y0 = p2[V_ + f0], y1 = p2[V_ + f1], y2 = p2[V_ + f2];
  float l[9];
  bary_coeffs(x0, x1, x2, y0, y1, y2, l);
  float px = (float)(pix & (W_ - 1));
  float py = (float)(pix >> 7);
  const float* cb = color + b * C_ * V_;
  float* img = out + (size_t)b * C_ * H_ * W_;
#pragma unroll
  for (int ch = 0; ch < C_; ++ch) {
    float c0 = cb[ch * V_ + f0], c1 = cb[ch * V_ + f1], c2 = cb[ch * V_ + f2];
    float Cx = c0 * l[0] + c1 * l[3] + c2 * l[6];
    float Cy = c0 * l[1] + c1 * l[4] + c2 * l[7];
    float Cc = c0 * l[2] + c1 * l[5] + c2 * l[8];
    img[ch * H_ * W_ + pix] = m * (Cx * px + Cy * py + Cc);
  }
  out[(size_t)B_ * C_ * H_ * W_ + (size_t)b * H_ * W_ + pix] = m;
}

// Phase 2: WMMA raster (planes x pixels) + fused argmin z-buffer + shading.
// One wave of 32 lanes handles 32 pixels (2 pixel-groups of 16).
__global__ void __launch_bounds__(128) raster_shade(const float* __restrict__ pt2d,
                                                    const float* __restrict__ color,
                                                    const int* __restrict__ face,
                                                    const float* __restrict__ planes,
                                                    float* __restrict__ out) {
  int wave = blockIdx.x * (blockDim.x >> 5) + (threadIdx.x >> 5);
  int lane = threadIdx.x & 31;
  int b = wave >> 9;          // 512 tiles (of 32 px) per batch
  int tile = wave & 511;
  int basePix = tile * 32;
  bool lo = lane < 16;
  int col = lane & 15;

  // B matrix (4x16, K rows = {x, y, 1, 0}). Layout mirrors A:
  // vgpr0: lanes0-15 = K0 (x), lanes16-31 = K2 (1); vgpr1: K1 (y) / K3 (0).
  v2f Bm[2];
#pragma unroll
  for (int g = 0; g < 2; ++g) {
    int pg = basePix + g * 16 + col;
    Bm[g].x = lo ? (float)(pg & (W_ - 1)) : 1.0f;
    Bm[g].y = lo ? (float)(pg >> 7) : 0.0f;
  }

  float bd[2] = {kINF, kINF};
  int bi[2] = {0, 0};

  // A matrix (16x4): lane m holds (cx,cy) [K0,K1], lane m+16 holds (cc,0) [K2,K3]
  const float2* abase = (const float2*)(planes + (size_t)b * F_ * 16) +
                        ((lane & 15) * 2 + (lo ? 0 : 1));
  int laneFace = lo ? 0 : 2;   // lanes0-15: faces fg*4+{0,1}; lanes16-31: +{2,3}
  v8f zero = {};

  // Software-pipelined A tile: load fg+1 before consuming fg; prefetch 8 ahead.
  float2 cur = abase[0];
  for (int fg = 0; fg < NFG_; ++fg) {
    int nfg = (fg + 1 < NFG_) ? (fg + 1) : (NFG_ - 1);
    int pfg = (fg + 8 < NFG_) ? (fg + 8) : (NFG_ - 1);
    float2 nxt = abase[nfg * 32];
    __builtin_prefetch(&abase[pfg * 32], 0, 3);
    v2f a; a.x = cur.x; a.y = cur.y;
    int fa = fg * 4 + laneFace;
#pragma unroll
    for (int g = 0; g < 2; ++g) {
      v8f d = __builtin_amdgcn_wmma_f32_16x16x4_f32(false, a, false, Bm[g],
                                                    (short)0, zero, false, false);
      // this lane's face 0: rows M%8 = 0..3 -> l0,l1,l2,depth  (min3 coverage test)
      float cand1 = (fminf(fminf(d[0], d[1]), d[2]) >= 0.0f) ? d[3] : kINF;
      bool t1 = cand1 < bd[g];
      bd[g] = t1 ? cand1 : bd[g];
      bi[g] = t1 ? fa : bi[g];
      // this lane's face 1: rows M%8 = 4..7
      float cand2 = (fminf(fminf(d[4], d[5]), d[6]) >= 0.0f) ? d[7] : kINF;
      bool t2 = cand2 < bd[g];
      bd[g] = t2 ? cand2 : bd[g];
      bi[g] = t2 ? (fa + 1) : bi[g];
    }
    cur = nxt;
  }

  // Merge the two face-halves (lane L <-> L+16); tie -> smaller face index (argmin-first).
#pragma unroll
  for (int g = 0; g < 2; ++g) {
    float od = __shfl_xor(bd[g], 16);
    int oi = __shfl_xor(bi[g], 16);
    bool take = (od < bd[g]) || ((od == bd[g]) && (oi < bi[g]));
    bd[g] = take ? od : bd[g];
    bi[g] = take ? oi : bi[g];
  }

  // Each lane shades 1 pixel: lanes0-15 -> group 0, lanes16-31 -> group 1.
  float sd = lo ? bd[0] : bd[1];
  int si = lo ? bi[0] : bi[1];
  int pix = basePix + lane;   // group*16 + col == lane
  shade_pixel(b, pix, sd, si, pt2d, color, face, out);
}

extern "C" void kernel_launch(void* const* d_in, const int* in_sizes, int n_in,
                              void* d_out, int out_size, void* d_ws, size_t ws_size,
                              hipStream_t stream) {
  const float* pt2d = (const float*)d_in[0];
  const float* color = (const float*)d_in[1];
  const float* pt3d = (const float*)d_in[2];
  const float* normal = (const float*)d_in[3];
  const float* Rm = (const float*)d_in[4];
  const float* Tm = (const float*)d_in[5];
  const int* face = (const int*)d_in[6];
  float* out = (float*)d_out;
  float* planes = (float*)d_ws;  // B*F*16 floats = 256 KB

  plane_setup<<<(B_ * F_ + 255) / 256, 256, 0, stream>>>(pt2d, pt3d, normal, Rm, Tm,
                                                         face, planes);
  // 1024 waves total (2 batches * 512 tiles of 32 px), 4 waves per block
  raster_shade<<<(B_ * 512) / 4, 128, 0, stream>>>(pt2d, color, face, planes, out);
}